// Net_53687091200002
// MI455X (gfx1250) — compile-verified
//
#include <hip/hip_runtime.h>

// SNN forward (snntorch Leaky, subtract reset), fused over all T=256 steps.
// One wave32 owns 16 batch rows for the whole time loop; fc1 runs on
// v_wmma_f32_16x16x4_f32 with mem1 held in the C/D register layout.
// Key algebraic fusion: reset(t) == spike(t-1), so the spike floats are
// carried across iterations and reused as the reset term -- one compare
// per hidden element per step instead of two.

typedef float v2f __attribute__((ext_vector_type(2)));
typedef float v8f __attribute__((ext_vector_type(8)));

#define SNN_BETA 0.95f
#define SNN_THR  1.0f
#define SNN_T    256
#define SNN_NI   23
#define SNN_NH   50

__global__ __launch_bounds__(128) void snn_fused_kernel(
    const float* __restrict__ x,    // [B, T, 23]
    const float* __restrict__ w1,   // [23, 50]
    const float* __restrict__ b1,   // [50]
    const float* __restrict__ w2,   // [50, 2]
    const float* __restrict__ b2,   // [2]
    float* __restrict__ out,        // spk2 [B,T,2] then mem2 [B,T,2]
    int Bsz)
{
  const int lane = threadIdx.x & 31;
  const int wave = threadIdx.x >> 5;
  const int b0   = (blockIdx.x * 4 + wave) * 16;   // 16 batch rows per wave
  const int half = lane >> 4;                      // 0: lanes 0-15, 1: lanes 16-31
  const int idx  = lane & 15;                      // N index (B/D), M index (A)
  const int kofs = 2 * half;                       // K sub-offset within a K=4 group

  // ---- preload w1 as B fragments: bf[k][j], k=0..5 (K=24 pad), j=0..3 (N=64 pad)
  v2f bf[6][4];
  float b1c[4];
  v2f w2r[4];
#pragma unroll
  for (int j = 0; j < 4; ++j) {
    const int col = j * 16 + idx;
    const bool cok = (col < SNN_NH);
    b1c[j] = cok ? b1[col] : 0.0f;
    if (cok) { w2r[j].x = w2[col * 2 + 0]; w2r[j].y = w2[col * 2 + 1]; }
    else     { w2r[j].x = 0.0f;            w2r[j].y = 0.0f; }
#pragma unroll
    for (int k = 0; k < 6; ++k) {
      const int r0 = 4 * k + kofs;
      v2f b;
      b.x = (cok && (r0     < SNN_NI)) ? w1[(r0    ) * SNN_NH + col] : 0.0f;
      b.y = (cok && (r0 + 1 < SNN_NI)) ? w1[(r0 + 1) * SNN_NH + col] : 0.0f;
      bf[k][j] = b;
    }
  }
  const float b2x = b2[0], b2y = b2[1];

  // A-source: lane holds batch row m = idx
  const float* xrow = x + (size_t)(b0 + idx) * (SNN_T * SNN_NI);

  // persistent state: mem1 (C/D layout) and spike floats (reset source)
  v8f mem1[4], sp[4];
#pragma unroll
  for (int j = 0; j < 4; ++j)
#pragma unroll
    for (int r = 0; r < 8; ++r) { mem1[j][r] = 0.0f; sp[j][r] = 0.0f; }

  const int rmine = idx & 7;            // mem2 row this lane tracks (within half)
  float mem2a = 0.0f, mem2b = 0.0f;     // mem2[rmine][0], mem2[rmine][1]
  float s2x = 0.0f, s2y = 0.0f;         // spk2 floats, reused as reset2 next step

  float* spkOut = out;
  float* memOut = out + (size_t)Bsz * SNN_T * 2;
  const size_t obase = (size_t)(b0 + 8 * half + rmine) * SNN_T * 2;

  for (int t = 0; t < SNN_T; ++t) {
    const float* xr = xrow + t * SNN_NI;

    // ---- A fragments: a[k] = { x[m, 4k+kofs], x[m, 4k+kofs+1] }, K padded to 24
    v2f a[6];
#pragma unroll
    for (int k = 0; k < 5; ++k) { a[k].x = xr[4 * k + kofs]; a[k].y = xr[4 * k + kofs + 1]; }
    a[5].x = xr[20 + kofs];
    a[5].y = half ? 0.0f : xr[21];      // element 23 is zero pad
    if (t + 8 < SNN_T) __builtin_prefetch(xr + 8 * SNN_NI, 0, 0);

    // ---- fc1 + leaky update fused into the WMMA accumulator chain:
    // C_seed = beta*mem1 + b1 - spk_prev*THR ; D = C_seed + x@w1 = mem1_new
#pragma unroll
    for (int j = 0; j < 4; ++j) {
      v8f c;
#pragma unroll
      for (int r = 0; r < 8; ++r)
        c[r] = __builtin_fmaf(SNN_BETA, mem1[j][r], b1c[j] - sp[j][r]);
#pragma unroll
      for (int k = 0; k < 6; ++k)
        c = __builtin_amdgcn_wmma_f32_16x16x4_f32(
                false, a[k], false, bf[k][j], (short)0, c, false, false);
      mem1[j] = c;
    }

    // ---- spikes (stored for next-step reset) + cur2 partials:
    // partial[2*r + o] = sum_j spk1[row r][col j*16+n] * w2[col][o]
    float partial[16];
#pragma unroll
    for (int i = 0; i < 16; ++i) partial[i] = 0.0f;
#pragma unroll
    for (int j = 0; j < 4; ++j) {
#pragma unroll
      for (int r = 0; r < 8; ++r) {
        const float sf = (mem1[j][r] > SNN_THR) ? 1.0f : 0.0f;
        sp[j][r] = sf;
        partial[2 * r + 0] = __builtin_fmaf(sf, w2r[j].x, partial[2 * r + 0]);
        partial[2 * r + 1] = __builtin_fmaf(sf, w2r[j].y, partial[2 * r + 1]);
      }
    }

    // ---- 16-lane xor butterfly (stays within each half); results replicated
#pragma unroll
    for (int m = 1; m <= 8; m <<= 1) {
#pragma unroll
      for (int i = 0; i < 16; ++i)
        partial[i] += __shfl_xor(partial[i], m, 32);
    }

    // ---- mem2 leaky update for this lane's row; reset2 = spk2(t-1)
    const float c2x = partial[2 * rmine + 0] + b2x;
    const float c2y = partial[2 * rmine + 1] + b2y;
    mem2a = __builtin_fmaf(SNN_BETA, mem2a, c2x - s2x);
    mem2b = __builtin_fmaf(SNN_BETA, mem2b, c2y - s2y);
    s2x = (mem2a > SNN_THR) ? 1.0f : 0.0f;
    s2y = (mem2b > SNN_THR) ? 1.0f : 0.0f;

    // ---- packed 8B stores: lanes idx<8 write spk2, idx>=8 write mem2
    float2 vs;
    float* dst;
    if (idx < 8) { vs.x = s2x;   vs.y = s2y;   dst = spkOut + obase + (size_t)t * 2; }
    else         { vs.x = mem2a; vs.y = mem2b; dst = memOut + obase + (size_t)t * 2; }
    *(float2*)dst = vs;
  }
}

extern "C" void kernel_launch(void* const* d_in, const int* in_sizes, int n_in,
                              void* d_out, int out_size, void* d_ws, size_t ws_size,
                              hipStream_t stream) {
  (void)n_in; (void)out_size; (void)d_ws; (void)ws_size;
  const float* x  = (const float*)d_in[0];
  const float* w1 = (const float*)d_in[1];
  const float* b1 = (const float*)d_in[2];
  const float* w2 = (const float*)d_in[3];
  const float* b2 = (const float*)d_in[4];
  float* out = (float*)d_out;

  const int Bsz = in_sizes[0] / (SNN_T * SNN_NI);   // 16384
  const int wavesPerBlock = 4;                      // 128 threads
  const int blocks = Bsz / (16 * wavesPerBlock);    // 256 blocks, 1024 waves
  snn_fused_kernel<<<blocks, 32 * wavesPerBlock, 0, stream>>>(
      x, w1, b1, w2, b2, out, Bsz);
}